// GraphDecoder_86595130622506
// MI455X (gfx1250) — compile-verified
//
#include <hip/hip_runtime.h>

#define DIMC 128
#define STATS_BLOCKS 256

typedef __attribute__((ext_vector_type(2))) float v2f;
typedef __attribute__((ext_vector_type(8))) float v8f;

// ---------------- degree / normalization prep ----------------

__global__ __launch_bounds__(256) void gd_init_deg(float* __restrict__ deg, int n) {
    int i = blockIdx.x * 256 + threadIdx.x;
    if (i < n) deg[i] = 1.0f;  // implicit self-loop weight
}

__global__ __launch_bounds__(256) void gd_deg_scatter(const int* __restrict__ dst,
                                                      const float* __restrict__ w,
                                                      float* __restrict__ deg, int e) {
    int i = blockIdx.x * 256 + threadIdx.x;
    if (i < e) atomicAdd(deg + dst[i], w[i]);
}

__global__ __launch_bounds__(256) void gd_norm(const float* __restrict__ deg,
                                               float* __restrict__ dinv,
                                               float* __restrict__ selfc, int n) {
    int i = blockIdx.x * 256 + threadIdx.x;
    if (i < n) {
        float d = deg[i];
        dinv[i] = rsqrtf(d);
        selfc[i] = 1.0f / d;
    }
}

__global__ __launch_bounds__(256) void gd_enorm(const int* __restrict__ src,
                                                const int* __restrict__ dst,
                                                const float* __restrict__ w,
                                                const float* __restrict__ dinv,
                                                float* __restrict__ enorm, int e) {
    int i = blockIdx.x * 256 + threadIdx.x;
    if (i < e) enorm[i] = dinv[dst[i]] * w[i] * dinv[src[i]];
}

// ---------------- fused GEMM (WMMA f32 16x16x4) + self-loop/bias init ----------------
// Block = 256 threads = 8 waves; each wave computes a 16-row x 128-col strip.
// W (128x128 f32, 64KB) staged in LDS once per block.
// Epilogue: xw[n][c] = (x@W)[n][c];  h[n][c] = selfc[n]*xw[n][c] + b[c]

__global__ __launch_bounds__(256) void gd_gemm_wmma(const float* __restrict__ x,
                                                    const float* __restrict__ W,
                                                    const float* __restrict__ b,
                                                    const float* __restrict__ selfc,
                                                    float* __restrict__ xw,
                                                    float* __restrict__ h, int nNodes) {
    __shared__ float Wlds[DIMC * DIMC];
    const int tid = threadIdx.x;
    // cooperative 64KB load of W into LDS
    for (int i = tid; i < DIMC * DIMC / 4; i += 256) {
        ((float4*)Wlds)[i] = ((const float4*)W)[i];
    }
    __syncthreads();

    const int wave = tid >> 5;          // 0..7
    const int lane = tid & 31;
    const int half = lane >> 4;         // 0/1
    const int l16  = lane & 15;
    const int rowBase = blockIdx.x * 128 + wave * 16;
    if (rowBase >= nNodes) return;      // wave-uniform exit (after last barrier)

    // A-frag row for this lane (clamped so OOB rows load valid memory; stores guarded)
    int aRow = rowBase + l16;
    if (aRow >= nNodes) aRow = nNodes - 1;
    const float* __restrict__ xrow = x + (size_t)aRow * DIMC;

    v8f acc[8] = {};  // 8 N-tiles of 16 -> full 128-wide output strip

    for (int k0 = 0; k0 < DIMC; k0 += 4) {
        v2f a;
        a.x = xrow[k0 + 2 * half];
        a.y = xrow[k0 + 2 * half + 1];
#pragma unroll
        for (int nt = 0; nt < 8; ++nt) {
            v2f bf;
            bf.x = Wlds[(k0 + 2 * half) * DIMC + nt * 16 + l16];
            bf.y = Wlds[(k0 + 2 * half + 1) * DIMC + nt * 16 + l16];
            acc[nt] = __builtin_amdgcn_wmma_f32_16x16x4_f32(
                false, a, false, bf, (short)0, acc[nt], false, false);
        }
    }

    // Epilogue: D layout VGPR r -> row r + 8*half, col = nt*16 + l16
#pragma unroll
    for (int r = 0; r < 8; ++r) {
        const int m = rowBase + r + 8 * half;
        if (m < nNodes) {
            const float sc = selfc[m];
#pragma unroll
            for (int nt = 0; nt < 8; ++nt) {
                const int c = nt * 16 + l16;
                const float v = acc[nt][r];
                const size_t idx = (size_t)m * DIMC + c;
                xw[idx] = v;
                h[idx] = sc * v + b[c];
            }
        }
    }
}

// ---------------- edge gather / scale / scatter-add ----------------
// 32 threads per edge, 4 channels (one float4) per thread.

__global__ __launch_bounds__(256) void gd_edge_scatter(const int* __restrict__ src,
                                                       const int* __restrict__ dst,
                                                       const float* __restrict__ enorm,
                                                       const float* __restrict__ xw,
                                                       float* __restrict__ h, int nEdges) {
    const long long t = (long long)blockIdx.x * 256 + threadIdx.x;
    const int e = (int)(t >> 5);
    if (e >= nEdges) return;
    const int c4 = ((int)t & 31) * 4;
    const float w = enorm[e];
    const int s = src[e], d = dst[e];
    const float4 v = *(const float4*)(xw + (size_t)s * DIMC + c4);
    float* hp = h + (size_t)d * DIMC + c4;
    atomicAdd(hp + 0, w * v.x);
    atomicAdd(hp + 1, w * v.y);
    atomicAdd(hp + 2, w * v.z);
    atomicAdd(hp + 3, w * v.w);
}

// ---------------- BatchNorm (training stats over nodes), deterministic 2-pass ----------------
// v = (relu? max(h,0):h) + latent ; stats per channel.

__global__ __launch_bounds__(128) void gd_bn_stats(const float* __restrict__ h,
                                                   const float* __restrict__ latent,
                                                   float* __restrict__ partial,
                                                   int nNodes, int relu) {
    const int c = threadIdx.x;  // 0..127
    const int chunk = (nNodes + (int)gridDim.x - 1) / (int)gridDim.x;
    const int n0 = blockIdx.x * chunk;
    int n1 = n0 + chunk; if (n1 > nNodes) n1 = nNodes;
    float s = 0.f, ss = 0.f;
    for (int n = n0; n < n1; ++n) {
        float v = h[(size_t)n * DIMC + c];
        if (relu) v = fmaxf(v, 0.f);
        v += latent[(size_t)n * DIMC + c];
        s += v; ss += v * v;
    }
    partial[(size_t)blockIdx.x * (2 * DIMC) + c] = s;
    partial[(size_t)blockIdx.x * (2 * DIMC) + DIMC + c] = ss;
}

__global__ __launch_bounds__(128) void gd_bn_finalize(const float* __restrict__ partial,
                                                      const float* __restrict__ gamma,
                                                      const float* __restrict__ beta,
                                                      float* __restrict__ sAB,
                                                      int nBlocks, int nNodes) {
    const int c = threadIdx.x;
    float s = 0.f, ss = 0.f;
    for (int bb = 0; bb < nBlocks; ++bb) {
        s  += partial[(size_t)bb * (2 * DIMC) + c];
        ss += partial[(size_t)bb * (2 * DIMC) + DIMC + c];
    }
    const float invN = 1.0f / (float)nNodes;
    const float mean = s * invN;
    const float var = ss * invN - mean * mean;  // biased var
    const float is = rsqrtf(var + 1e-5f);
    const float g = gamma[c] * is;
    sAB[c] = g;                       // scale
    sAB[DIMC + c] = beta[c] - g * mean;  // shift
}

__global__ __launch_bounds__(256) void gd_bn_apply(const float* __restrict__ h,
                                                   const float* __restrict__ latent,
                                                   const float* __restrict__ sAB,
                                                   float* __restrict__ out,
                                                   int nNodes, int relu) {
    const size_t t = (size_t)blockIdx.x * 256 + threadIdx.x;
    if (t >= (size_t)nNodes * DIMC) return;
    const int c = (int)(t & (DIMC - 1));
    float v = h[t];
    if (relu) v = fmaxf(v, 0.f);
    v += latent[t];
    out[t] = sAB[c] * v + sAB[DIMC + c];
}

// ---------------- launch ----------------

extern "C" void kernel_launch(void* const* d_in, const int* in_sizes, int n_in,
                              void* d_out, int out_size, void* d_ws, size_t ws_size,
                              hipStream_t stream) {
    const float* latent = (const float*)d_in[0];
    const int*   ei     = (const int*)d_in[1];   // [2, E] flat: src then dst
    const float* ew     = (const float*)d_in[2];
    const float* Ws     = (const float*)d_in[3]; // [3,128,128]
    const float* bs     = (const float*)d_in[4]; // [3,128]
    const float* gammas = (const float*)d_in[5];
    const float* betas  = (const float*)d_in[6];

    const int N = in_sizes[0] / DIMC;
    const int E = in_sizes[2];
    const int* src = ei;
    const int* dst = ei + E;

    // workspace carve-out (256B aligned)
    char* p = (char*)d_ws;
    auto alloc = [&](size_t bytes) -> char* {
        char* r = p;
        p += (bytes + 255) & ~(size_t)255;
        return r;
    };
    float* deg     = (float*)alloc((size_t)N * 4);
    float* dinv    = (float*)alloc((size_t)N * 4);
    float* selfc   = (float*)alloc((size_t)N * 4);
    float* enorm   = (float*)alloc((size_t)E * 4);
    float* xw      = (float*)alloc((size_t)N * DIMC * 4);
    float* hbuf    = (float*)alloc((size_t)N * DIMC * 4);
    float* xbuf    = (float*)alloc((size_t)N * DIMC * 4);
    float* partial = (float*)alloc((size_t)STATS_BLOCKS * 2 * DIMC * 4);
    float* sAB     = (float*)alloc(2 * DIMC * 4);

    const int gN   = (N + 255) / 256;
    const int gE   = (E + 255) / 256;
    const int gGem = (N + 127) / 128;
    const int gSc  = (int)(((long long)E * 32 + 255) / 256);
    const int gNC  = (int)(((size_t)N * DIMC + 255) / 256);

    // one-time graph normalization
    gd_init_deg   <<<gN, 256, 0, stream>>>(deg, N);
    gd_deg_scatter<<<gE, 256, 0, stream>>>(dst, ew, deg, E);
    gd_norm       <<<gN, 256, 0, stream>>>(deg, dinv, selfc, N);
    gd_enorm      <<<gE, 256, 0, stream>>>(src, dst, ew, dinv, enorm, E);

    const float* xin = latent;
    for (int i = 0; i < 3; ++i) {
        const int relu = (i != 1) ? 1 : 0;  // i != N_LAYERS-2
        gd_gemm_wmma   <<<gGem, 256, 0, stream>>>(xin, Ws + (size_t)i * DIMC * DIMC,
                                                  bs + (size_t)i * DIMC, selfc, xw, hbuf, N);
        gd_edge_scatter<<<gSc, 256, 0, stream>>>(src, dst, enorm, xw, hbuf, E);
        gd_bn_stats    <<<STATS_BLOCKS, 128, 0, stream>>>(hbuf, latent, partial, N, relu);
        gd_bn_finalize <<<1, 128, 0, stream>>>(partial, gammas + (size_t)i * DIMC,
                                               betas + (size_t)i * DIMC, sAB, STATS_BLOCKS, N);
        float* dstbuf = (i == 2) ? (float*)d_out : xbuf;
        gd_bn_apply    <<<gNC, 256, 0, stream>>>(hbuf, latent, sAB, dstbuf, N, relu);
        xin = dstbuf;
    }
}